// loss_39934605918651
// MI455X (gfx1250) — compile-verified
//
#include <hip/hip_runtime.h>

// ---------------------------------------------------------------------------
// Loss kernel for MI455X (gfx1250, wave32).
//
// Work decomposition: B*36 = 9216 items; one block of 1024 threads (32 waves,
// one full WGP), 9 items per thread. gt_pos staged in LDS; the cross-batch
// sq term collapses to 4 integer moments (Sx, Sy, Sxx, Syy) computed exactly
// with deterministic LDS integer atomics. Final 1024->1 reduction done on
// wave 0 via chained V_WMMA_F32_16X16X4_F32 (B = ones, C accumulates).
// ---------------------------------------------------------------------------

typedef float v2f __attribute__((ext_vector_type(2)));
typedef float v8f __attribute__((ext_vector_type(8)));

#define S_DIM 1024
#define B_DIM 256
#define NITEMS (B_DIM * 36)  // 9216

__global__ __launch_bounds__(1024) void loss_39934605918651_kernel(
    const float* __restrict__ yp,   // [256][1024][1024]
    const int*   __restrict__ gt,   // [256][2]
    float*       __restrict__ out)  // [1]
{
    __shared__ int   s_gt[2 * B_DIM];   // staged gt_pos
    __shared__ int   s_sums[4];         // Sx, Sy, Sxx, Syy (exact int)
    __shared__ int   s_cnt;             // number of valid positions
    __shared__ float s_part[1024];      // per-thread partial sums

    const int t = threadIdx.x;

    if (t < 2 * B_DIM) s_gt[t] = gt[t];
    if (t < 4)  s_sums[t] = 0;
    if (t == 0) s_cnt = 0;
    __syncthreads();

    // Exact integer moments of gt_pos (deterministic: integer LDS atomics).
    if (t < B_DIM) {
        int gx = s_gt[2 * t];
        int gy = s_gt[2 * t + 1];
        atomicAdd(&s_sums[0], gx);
        atomicAdd(&s_sums[1], gy);
        atomicAdd(&s_sums[2], gx * gx);   // < 2.7e8, fits int32
        atomicAdd(&s_sums[3], gy * gy);
    }
    __syncthreads();

    const long long Sx  = s_sums[0];
    const long long Sy  = s_sums[1];
    const long long Sxx = s_sums[2];
    const long long Syy = s_sums[3];

    float part = 0.0f;
    int   cnt  = 0;

    #pragma unroll
    for (int it = 0; it < NITEMS / 1024; ++it) {
        const int idx = t + it * 1024;          // 0..9215
        const int b   = idx / 36;
        const int k   = idx - b * 36;
        const int oi  = k / 6 - 3;              // offsets in [-3, 2]
        const int oj  = (k - (k / 6) * 6) - 3;

        const int gx = s_gt[2 * b];
        const int gy = s_gt[2 * b + 1];
        const int px = gx + oi;
        const int py = gy + oj;

        const bool valid = (px >= 0) & (px < S_DIM) & (py >= 0) & (py < S_DIM);
        if (valid) {
            // valid => clip is identity
            const float y_hat =
                yp[((size_t)b << 20) + ((size_t)px << 10) + (size_t)py];

            // sq = sum_{b'} (px-gx[b'])^2 + (py-gy[b'])^2, exact in int64:
            const long long sq =
                (long long)B_DIM * ((long long)px * px + (long long)py * py)
                - 2LL * ((long long)px * Sx + (long long)py * Sy)
                + Sxx + Syy;

            const float y = expf(-(float)sq / 5.0f);   // underflows to 0.0f
            const float d = y_hat - y;
            float term;
            if (y == 1.0f) {
                term = -logf(y_hat) * d * d;
            } else {
                const float om  = 1.0f - y;
                const float om2 = om * om;
                term = -logf(1.0f - y_hat) * (om2 * om2) * d * d;
            }
            part += term;
            cnt  += 1;
        }
    }

    s_part[t] = part;
    atomicAdd(&s_cnt, cnt);
    __syncthreads();

    // ---- terminal 1024 -> 1 reduction on wave 0 -------------------------
    if (t < 32) {   // whole wave takes the branch -> EXEC all ones for WMMA
#if __has_builtin(__builtin_amdgcn_wmma_f32_16x16x4_f32)
        v8f c = {0.f, 0.f, 0.f, 0.f, 0.f, 0.f, 0.f, 0.f};
        v2f bones;
        bones[0] = 1.0f;
        bones[1] = 1.0f;
        // 16 chained WMMAs: each consumes 64 partials (any A-layout bijection
        // sums correctly because B is all-ones); C accumulates.
        #pragma unroll
        for (int i = 0; i < 16; ++i) {
            v2f a;
            a[0] = s_part[i * 64 + 2 * t];
            a[1] = s_part[i * 64 + 2 * t + 1];
            c = __builtin_amdgcn_wmma_f32_16x16x4_f32(
                    false, a, false, bones, (short)0, c, false, false);
        }
        // D[m,n] identical across columns n. Lane 0 holds D[0..7,0] in c[0..7],
        // lane 16 holds D[8..15,0]. Total = lane0 sum + lane16 sum.
        float s = c[0] + c[1] + c[2] + c[3] + c[4] + c[5] + c[6] + c[7];
        float s_hi = __shfl(s, 16, 32);
        if (t == 0) out[0] = (s + s_hi) / (float)s_cnt;
#else
        // Fallback: shuffle tree reduction.
        float s = 0.0f;
        for (int i = t; i < 1024; i += 32) s += s_part[i];
        for (int off = 16; off > 0; off >>= 1) s += __shfl_down(s, off, 32);
        if (t == 0) out[0] = s / (float)s_cnt;
#endif
    }
}

extern "C" void kernel_launch(void* const* d_in, const int* in_sizes, int n_in,
                              void* d_out, int out_size, void* d_ws, size_t ws_size,
                              hipStream_t stream) {
    const float* yp = (const float*)d_in[0];   // y_predict: 256*1024*1024 f32
    const int*   gt = (const int*)d_in[1];     // gt_pos:    256*2 i32
    float*       out = (float*)d_out;          // scalar f32

    hipLaunchKernelGGL(loss_39934605918651_kernel,
                       dim3(1), dim3(1024), 0, stream, yp, gt, out);
}